// GCNClassifier_72215580115747
// MI455X (gfx1250) — compile-verified
//
#include <hip/hip_runtime.h>
#include <hip/hip_bf16.h>
#include <math.h>

typedef __attribute__((ext_vector_type(2))) float v2f;
typedef __attribute__((ext_vector_type(8))) float v8f;

#define D 128          // feature dim (d_in == d_h == 128)

// ---------------------------------------------------------------------------
// Degree / normalization kernels
// ---------------------------------------------------------------------------
__global__ void k_init_deg(float* deg, int n) {
    int i = blockIdx.x * blockDim.x + threadIdx.x;
    if (i < n) deg[i] = 1.0f;                 // self loop
}

__global__ void k_count_deg(const int* __restrict__ ei, int E, float* deg) {
    int e = blockIdx.x * blockDim.x + threadIdx.x;
    if (e < E) atomicAdd(&deg[ei[E + e]], 1.0f);   // col = target
}

__global__ void k_dinv(const float* __restrict__ deg, float* dinv, int n) {
    int i = blockIdx.x * blockDim.x + threadIdx.x;
    if (i < n) {
        float d = deg[i];
        dinv[i] = (d > 0.0f) ? rsqrtf(d) : 0.0f;
    }
}

// ---------------------------------------------------------------------------
// GEMM: C[n x 128] = A[n x 128] @ W[128 x 128] using V_WMMA_F32_16X16X4_F32.
// One block = 256 threads = 8 waves; each wave owns a 16x128 row tile.
// W lives in LDS (64 KB). 32 K-steps x 8 N-tiles = 256 WMMA per wave.
// Out-of-range rows are CLAMPED on load (branch-free inner loop); their
// products land only in output rows that the guarded store discards.
// ---------------------------------------------------------------------------
__global__ __launch_bounds__(256) void k_gemm_wmma(
        const float* __restrict__ A, const float* __restrict__ W,
        float* __restrict__ C, int n) {
    __shared__ float Wl[D * D];

    // cooperative load of W into LDS (float4, fully coalesced)
    for (int i = threadIdx.x; i < (D * D) / 4; i += 256)
        ((float4*)Wl)[i] = ((const float4*)W)[i];
    __syncthreads();

    const int wave = threadIdx.x >> 5;
    const int lane = threadIdx.x & 31;
    const int m0   = blockIdx.x * 128 + wave * 16;

    int arow = m0 + (lane & 15);               // A: both half-waves share rows
    if (arow > n - 1) arow = n - 1;            // clamp: branch-free loads
    const int khalf = (lane >> 4) * 2;         // lanes>=16 hold K pair {2,3}
    const float* Ar = A + (size_t)arow * D;

    v8f acc[8] = {};                           // 8 N-tiles of 16x16 f32 accum

    for (int k = 0; k < D; k += 4) {
        // (k + khalf) is even -> 8-byte aligned -> single global_load_b64
        const v2f a = *(const v2f*)(Ar + k + khalf);
        const int kb = k + khalf;
#pragma unroll
        for (int t = 0; t < 8; ++t) {
            const int col = t * 16 + (lane & 15);
            v2f b;
            b.x = Wl[kb * D + col];
            b.y = Wl[(kb + 1) * D + col];
            acc[t] = __builtin_amdgcn_wmma_f32_16x16x4_f32(
                false, a, false, b, (short)0, acc[t], false, false);
        }
    }

    // C/D layout: VGPR v -> M = v + 8*(lane>=16), N = lane&15 within tile
    const int rbase = m0 + (lane >> 4) * 8;
    const int cix   = lane & 15;
#pragma unroll
    for (int t = 0; t < 8; ++t) {
#pragma unroll
        for (int v = 0; v < 8; ++v) {
            const int r = rbase + v;
            if (r < n) C[(size_t)r * D + t * 16 + cix] = acc[t][v];
        }
    }
}

// ---------------------------------------------------------------------------
// out[i] = xw[i] * dinv[i]^2   (self-loop contribution; also initializes out)
// one thread per float4
// ---------------------------------------------------------------------------
__global__ void k_self_init(const float* __restrict__ xw,
                            const float* __restrict__ dinv,
                            float* __restrict__ out, int n) {
    size_t i = (size_t)blockIdx.x * blockDim.x + threadIdx.x;   // float4 index
    if (i >= (size_t)n * (D / 4)) return;
    int node = (int)(i >> 5);                                    // D/4 == 32
    float dd = dinv[node]; dd *= dd;
    float4 v = ((const float4*)xw)[i];
    float4 o = {v.x * dd, v.y * dd, v.z * dd, v.w * dd};
    ((float4*)out)[i] = o;
}

// ---------------------------------------------------------------------------
// Edge scatter: out[col] += xw[row] * dinv[row]*dinv[col].
// One wave per edge: 32 lanes x float4 = 128 coalesced floats.
// ---------------------------------------------------------------------------
__global__ __launch_bounds__(256) void k_agg(
        const int* __restrict__ ei, int E,
        const float* __restrict__ xw, const float* __restrict__ dinv,
        float* __restrict__ out) {
    int e = blockIdx.x * 8 + (threadIdx.x >> 5);
    if (e >= E) return;
    const int lane = threadIdx.x & 31;
    const int r = ei[e];
    const int c = ei[E + e];
    const float nrm = dinv[r] * dinv[c];
    float4 v = ((const float4*)(xw + (size_t)r * D))[lane];
    float* dst = out + (size_t)c * D + lane * 4;
    atomicAdd(dst + 0, v.x * nrm);
    atomicAdd(dst + 1, v.y * nrm);
    atomicAdd(dst + 2, v.z * nrm);
    atomicAdd(dst + 3, v.w * nrm);
}

// ---------------------------------------------------------------------------
// h = relu(h + b)
// ---------------------------------------------------------------------------
__global__ void k_bias_relu(float* h, const float* __restrict__ b, int n) {
    size_t i = (size_t)blockIdx.x * blockDim.x + threadIdx.x;
    if (i >= (size_t)n * D) return;
    int f = (int)(i & (D - 1));
    float v = h[i] + b[f];
    h[i] = v > 0.0f ? v : 0.0f;
}

// ---------------------------------------------------------------------------
// Edge classifier: concat(h[row], h[col]) @ Wfc + bfc, then log_softmax.
// One wave per edge; lane l handles features 4l..4l+3 of both halves,
// butterfly shuffle reduction over the wave (warpSize == 32 on gfx1250).
// ---------------------------------------------------------------------------
__global__ __launch_bounds__(256) void k_edge_cls(
        const int* __restrict__ ei, int E,
        const float* __restrict__ h, const float* __restrict__ Wfc,
        const float* __restrict__ bfc, float* __restrict__ out) {
    int e = blockIdx.x * 8 + (threadIdx.x >> 5);
    if (e >= E) return;
    const int lane = threadIdx.x & 31;
    const int r = ei[e];
    const int c = ei[E + e];
    float4 vr = ((const float4*)(h + (size_t)r * D))[lane];
    float4 vc = ((const float4*)(h + (size_t)c * D))[lane];
    const int k0 = lane * 4;
    float s0 = 0.f, s1 = 0.f, s2 = 0.f, s3 = 0.f;
    const float* pr = (const float*)&vr;
    const float* pc = (const float*)&vc;
#pragma unroll
    for (int i = 0; i < 4; ++i) {
        float a  = pr[i];
        float bb = pc[i];
        float4 wr = ((const float4*)Wfc)[k0 + i];        // Wfc[k, 0..3]
        float4 wc = ((const float4*)Wfc)[D + k0 + i];    // Wfc[128+k, 0..3]
        s0 += a * wr.x + bb * wc.x;
        s1 += a * wr.y + bb * wc.y;
        s2 += a * wr.z + bb * wc.z;
        s3 += a * wr.w + bb * wc.w;
    }
#pragma unroll
    for (int m = 16; m > 0; m >>= 1) {
        s0 += __shfl_xor(s0, m);
        s1 += __shfl_xor(s1, m);
        s2 += __shfl_xor(s2, m);
        s3 += __shfl_xor(s3, m);
    }
    if (lane == 0) {
        s0 += bfc[0]; s1 += bfc[1]; s2 += bfc[2]; s3 += bfc[3];
        float mx  = fmaxf(fmaxf(s0, s1), fmaxf(s2, s3));
        float sum = __expf(s0 - mx) + __expf(s1 - mx) +
                    __expf(s2 - mx) + __expf(s3 - mx);
        float lse = mx + __logf(sum);
        float4 o  = {s0 - lse, s1 - lse, s2 - lse, s3 - lse};
        ((float4*)out)[e] = o;
    }
}

// ---------------------------------------------------------------------------
// Launch
// ---------------------------------------------------------------------------
extern "C" void kernel_launch(void* const* d_in, const int* in_sizes, int n_in,
                              void* d_out, int out_size, void* d_ws, size_t ws_size,
                              hipStream_t stream) {
    const float* x   = (const float*)d_in[0];
    const int*   ei  = (const int*)  d_in[1];
    const float* W1  = (const float*)d_in[2];
    const float* b1  = (const float*)d_in[3];
    const float* W2  = (const float*)d_in[4];
    const float* b2  = (const float*)d_in[5];
    const float* Wfc = (const float*)d_in[6];
    const float* bfc = (const float*)d_in[7];
    float* out = (float*)d_out;

    const int n = in_sizes[0] / D;     // 50000
    const int E = in_sizes[1] / 2;     // 600000

    float* bufA = (float*)d_ws;                 // xW scratch
    float* bufB = bufA + (size_t)n * D;         // aggregated / h
    float* deg  = bufB + (size_t)n * D;
    float* dinv = deg + n;

    const int gN    = (n + 255) / 256;
    const int gE    = (E + 255) / 256;
    const int gGemm = (n + 127) / 128;
    const int gVec4 = (int)(((size_t)n * (D / 4) + 255) / 256);
    const int gFeat = (int)(((size_t)n * D + 255) / 256);
    const int gWave = (E + 7) / 8;              // wave-per-edge kernels

    // degrees + symmetric normalization
    k_init_deg <<<gN, 256, 0, stream>>>(deg, n);
    k_count_deg<<<gE, 256, 0, stream>>>(ei, E, deg);
    k_dinv     <<<gN, 256, 0, stream>>>(deg, dinv, n);

    // layer 1
    k_gemm_wmma<<<gGemm, 256, 0, stream>>>(x, W1, bufA, n);
    k_self_init<<<gVec4, 256, 0, stream>>>(bufA, dinv, bufB, n);
    k_agg      <<<gWave, 256, 0, stream>>>(ei, E, bufA, dinv, bufB);
    k_bias_relu<<<gFeat, 256, 0, stream>>>(bufB, b1, n);

    // layer 2
    k_gemm_wmma<<<gGemm, 256, 0, stream>>>(bufB, W2, bufA, n);
    k_self_init<<<gVec4, 256, 0, stream>>>(bufA, dinv, bufB, n);
    k_agg      <<<gWave, 256, 0, stream>>>(ei, E, bufA, dinv, bufB);
    k_bias_relu<<<gFeat, 256, 0, stream>>>(bufB, b2, n);

    // edge classifier + log_softmax
    k_edge_cls <<<gWave, 256, 0, stream>>>(ei, E, bufB, Wfc, bfc, out);
}